// MoERouter_83399674953936
// MI455X (gfx1250) — compile-verified
//
#include <hip/hip_runtime.h>
#include <math.h>
#include <stdint.h>

// ---------------------------------------------------------------------------
// MoE Router for MI455X (gfx1250, wave32)
//   logits = x[16384,4096] @ W[64,4096]^T + b   (fp32 WMMA 16x16x4, split-K)
//   probs  = softmax(logits) -> top-8 -> renormalized weights, one-hot mask
//
// HBM-bound: 256 MB x-read at 23.3 TB/s => ~11.5us floor. Split-K x4 raises
// wave count to 4096 for latency hiding without increasing x traffic; W is
// L2-resident (1 MB vs 192 MB L2) and staged into double-buffered LDS with
// CDNA5 async global->LDS copies overlapped with WMMA compute.
// ---------------------------------------------------------------------------

#define NTOK 16384
#define HID  4096
#define NEXP 64
#define TOPK 8

#define KT          128              // K-chunk staged in LDS for W
#define LDS_STRIDE  (KT + 4)         // 132 dwords: disjoint banks for both halves
#define WAVES_PB    4                // waves per block
#define TOK_PB      (WAVES_PB * 16)  // 64 tokens per block
#define NSPLIT      4                // split-K factor (needs 16 MB workspace)

typedef __attribute__((ext_vector_type(2))) float v2f;
typedef __attribute__((ext_vector_type(4))) float v4f;
typedef __attribute__((ext_vector_type(8))) float v8f;
typedef __attribute__((ext_vector_type(4))) int   v4i;

// address-space qualified element types for the async global->LDS builtin
typedef __attribute__((address_space(1))) v4i v4i_g;   // global int4
typedef __attribute__((address_space(3))) v4i v4i_l;   // LDS int4

#if defined(__has_builtin)
#  if __has_builtin(__builtin_amdgcn_global_load_async_to_lds_b128)
#    define USE_ASYNC_LDS 1
#  endif
#endif
#ifndef USE_ASYNC_LDS
#  define USE_ASYNC_LDS 0
#endif

__global__ __launch_bounds__(WAVES_PB * 32)
void moe_router_gemm(const float* __restrict__ x,
                     const float* __restrict__ W,
                     float* __restrict__ part,   // [gridDim.y][NTOK][NEXP]
                     int klen) {
  __shared__ __align__(16) float wlds[2][NEXP * LDS_STRIDE];

  const int tid   = threadIdx.x;
  const int wave  = tid >> 5;
  const int lane  = tid & 31;
  const int nn    = lane & 15;   // expert column within tile / token row for A
  const int khalf = lane >> 4;   // which K-pair this lane holds

  const int rowBase = blockIdx.x * TOK_PB + wave * 16;
  const int kBase   = blockIdx.y * klen;
  const int nchunk  = klen / KT;

  // A fragment source: row (rowBase + nn), columns k + khalf*2 + {0,1}
  const float* aptr = x + (size_t)(rowBase + nn) * HID + (khalf * 2);

  v8f acc[4];
  acc[0] = (v8f){}; acc[1] = (v8f){}; acc[2] = (v8f){}; acc[3] = (v8f){};

  int buf = 0;

  // ---------------- stage first W chunk ----------------
#if USE_ASYNC_LDS
  for (int i = tid; i < NEXP * (KT / 4); i += WAVES_PB * 32) {
    const int e  = i / (KT / 4);
    const int kq = (i % (KT / 4)) * 4;
    __builtin_amdgcn_global_load_async_to_lds_b128(
        (v4i_g*)(uintptr_t)(W + (size_t)e * HID + kBase + kq),
        (v4i_l*)(uintptr_t)&wlds[0][e * LDS_STRIDE + kq], 0, 0);
  }
  asm volatile("s_wait_asynccnt 0x0" ::: "memory");
  __syncthreads();
#else
  for (int i = tid; i < NEXP * (KT / 4); i += WAVES_PB * 32) {
    const int e  = i / (KT / 4);
    const int kq = (i % (KT / 4)) * 4;
    *(v4f*)(&wlds[0][e * LDS_STRIDE + kq]) =
        *(const v4f*)(W + (size_t)e * HID + kBase + kq);
  }
  __syncthreads();
#endif

  for (int c = 0; c < nchunk; ++c) {
    const int k0 = kBase + c * KT;

#if USE_ASYNC_LDS
    // issue async copies for next chunk into the other buffer (overlapped)
    if (c + 1 < nchunk) {
      const int kn = k0 + KT;
      for (int i = tid; i < NEXP * (KT / 4); i += WAVES_PB * 32) {
        const int e  = i / (KT / 4);
        const int kq = (i % (KT / 4)) * 4;
        __builtin_amdgcn_global_load_async_to_lds_b128(
            (v4i_g*)(uintptr_t)(W + (size_t)e * HID + kn + kq),
            (v4i_l*)(uintptr_t)&wlds[buf ^ 1][e * LDS_STRIDE + kq], 0, 0);
      }
    }
#endif
    // prefetch next chunk of x (covers all 4 cachelines of each row's 512 B)
    if (c + 1 < nchunk) {
      const float* pf = x + (size_t)(rowBase + nn) * HID + (k0 + KT) + khalf * 32;
      __builtin_prefetch(pf, 0, 0);
      __builtin_prefetch(pf + 64, 0, 0);
    }

    const float* wb = &wlds[buf][0];
#pragma unroll 8
    for (int kk = 0; kk < KT; kk += 4) {
      v2f a = *(const v2f*)(aptr + k0 + kk);
      const float* bb = &wb[nn * LDS_STRIDE + kk + khalf * 2];
      v2f b0 = *(const v2f*)(bb + 0 * 16 * LDS_STRIDE);
      v2f b1 = *(const v2f*)(bb + 1 * 16 * LDS_STRIDE);
      v2f b2 = *(const v2f*)(bb + 2 * 16 * LDS_STRIDE);
      v2f b3 = *(const v2f*)(bb + 3 * 16 * LDS_STRIDE);
      acc[0] = __builtin_amdgcn_wmma_f32_16x16x4_f32(false, a, false, b0,
                                                     (short)0, acc[0], false, false);
      acc[1] = __builtin_amdgcn_wmma_f32_16x16x4_f32(false, a, false, b1,
                                                     (short)0, acc[1], false, false);
      acc[2] = __builtin_amdgcn_wmma_f32_16x16x4_f32(false, a, false, b2,
                                                     (short)0, acc[2], false, false);
      acc[3] = __builtin_amdgcn_wmma_f32_16x16x4_f32(false, a, false, b3,
                                                     (short)0, acc[3], false, false);
    }

#if USE_ASYNC_LDS
    asm volatile("s_wait_asynccnt 0x0" ::: "memory");
    __syncthreads();
    buf ^= 1;
#else
    __syncthreads();
    if (c + 1 < nchunk) {
      const int kn = k0 + KT;
      for (int i = tid; i < NEXP * (KT / 4); i += WAVES_PB * 32) {
        const int e  = i / (KT / 4);
        const int kq = (i % (KT / 4)) * 4;
        *(v4f*)(&wlds[0][e * LDS_STRIDE + kq]) =
            *(const v4f*)(W + (size_t)e * HID + kn + kq);
      }
      __syncthreads();
    }
#endif
  }

  // ---- store partial: C/D layout: VGPR v, lane -> (m = v + khalf*8, n = nn) --
  float* p = part + (size_t)blockIdx.y * NTOK * NEXP;
#pragma unroll
  for (int t = 0; t < 4; ++t) {
#pragma unroll
    for (int v = 0; v < 8; ++v) {
      const int row = rowBase + khalf * 8 + v;
      p[(size_t)row * NEXP + t * 16 + nn] = acc[t][v];
    }
  }
}

// ---------------------------------------------------------------------------
// Finalize: split-K reduction + bias, softmax over 64 experts, top-8
// (lowest-index tie-break), renormalize, one-hot mask. One thread per token;
// probs stay in VGPRs (selection via bitmask, no dynamic array writes).
// NOTE: `part` may alias `logits` when nsplit==1 -> no __restrict__ on them.
// ---------------------------------------------------------------------------
__global__ __launch_bounds__(128)
void moe_router_finalize(const float* part, int nsplit,
                         const float* __restrict__ bias,
                         float* logits,
                         float* __restrict__ weights,
                         float* __restrict__ indices,
                         float* __restrict__ mask) {
  const int tok = blockIdx.x * blockDim.x + threadIdx.x;
  if (tok >= NTOK) return;

  float v[NEXP];
#pragma unroll
  for (int e = 0; e < NEXP; ++e) v[e] = bias[e];

  const size_t slab = (size_t)NTOK * NEXP;
  for (int s = 0; s < nsplit; ++s) {
    const float* p = part + (size_t)s * slab + (size_t)tok * NEXP;
#pragma unroll
    for (int e = 0; e < NEXP; ++e) v[e] += p[e];
  }
#pragma unroll
  for (int e = 0; e < NEXP; ++e) logits[(size_t)tok * NEXP + e] = v[e];

  // softmax
  float mx = -INFINITY;
#pragma unroll
  for (int e = 0; e < NEXP; ++e) mx = fmaxf(mx, v[e]);
  float sum = 0.f;
#pragma unroll
  for (int e = 0; e < NEXP; ++e) { v[e] = expf(v[e] - mx); sum += v[e]; }
  const float inv = 1.f / sum;
#pragma unroll
  for (int e = 0; e < NEXP; ++e) v[e] *= inv;

  // top-8, first-wins on ties (matches jax.lax.top_k)
  unsigned long long picked = 0ull;
  float tw[TOPK];
  int   ti[TOPK];
  float tsum = 0.f;
#pragma unroll
  for (int s = 0; s < TOPK; ++s) {
    float best = -1.f;
    int   bi   = 0;
#pragma unroll
    for (int e = 0; e < NEXP; ++e) {
      const bool avail = ((picked >> e) & 1ull) == 0ull;
      if (avail && v[e] > best) { best = v[e]; bi = e; }
    }
    tw[s] = best;
    ti[s] = bi;
    tsum += best;
    picked |= (1ull << bi);
  }

  const float wnorm = 1.f / (tsum + 1e-6f);
#pragma unroll
  for (int s = 0; s < TOPK; ++s) {
    weights[(size_t)tok * TOPK + s] = tw[s] * wnorm;
    indices[(size_t)tok * TOPK + s] = (float)ti[s];
  }
#pragma unroll
  for (int e = 0; e < NEXP; ++e)
    mask[(size_t)tok * NEXP + e] = ((picked >> e) & 1ull) ? 1.f : 0.f;
}

// ---------------------------------------------------------------------------
extern "C" void kernel_launch(void* const* d_in, const int* in_sizes, int n_in,
                              void* d_out, int out_size, void* d_ws, size_t ws_size,
                              hipStream_t stream) {
  const float* x = (const float*)d_in[0];   // [NTOK, HID]
  const float* W = (const float*)d_in[1];   // [NEXP, HID]
  const float* b = (const float*)d_in[2];   // [NEXP]

  float* out     = (float*)d_out;
  float* logits  = out;                                      // NTOK*NEXP
  float* weights = logits  + (size_t)NTOK * NEXP;            // NTOK*TOPK
  float* indices = weights + (size_t)NTOK * TOPK;            // NTOK*TOPK
  float* mask    = indices + (size_t)NTOK * TOPK;            // NTOK*NEXP

  const size_t need = (size_t)NSPLIT * NTOK * NEXP * sizeof(float);
  const int nsplit  = (ws_size >= need) ? NSPLIT : 1;        // deterministic
  float* part       = (nsplit == 1) ? logits : (float*)d_ws;
  const int klen    = HID / nsplit;

  dim3 gemmGrid(NTOK / TOK_PB, nsplit);    // 256 x nsplit blocks
  dim3 gemmBlock(WAVES_PB * 32);           // 128 threads = 4 waves
  moe_router_gemm<<<gemmGrid, gemmBlock, 0, stream>>>(x, W, part, klen);

  dim3 finGrid((NTOK + 127) / 128);
  dim3 finBlock(128);
  moe_router_finalize<<<finGrid, finBlock, 0, stream>>>(part, nsplit, b, logits,
                                                        weights, indices, mask);
}